// MemoryPool_88965952569956
// MI455X (gfx1250) — compile-verified
//
#include <hip/hip_runtime.h>
#include <hip/hip_bf16.h>
#include <math.h>

// ---------------------------------------------------------------------------
// MI455X (gfx1250). Memory-bound workload => f32 end-to-end via
// V_WMMA_F32_16X16X4_F32, with CDNA5 async global->LDS double-buffered
// staging (GLOBAL_LOAD_ASYNC_TO_LDS_B128 / ASYNCcnt) in the GEMM.
// ---------------------------------------------------------------------------

typedef float v2f __attribute__((ext_vector_type(2)));
typedef float v8f __attribute__((ext_vector_type(8)));

__device__ __forceinline__ v8f wmma_f32x4(v2f a, v2f b, v8f c) {
  return __builtin_amdgcn_wmma_f32_16x16x4_f32(false, a, false, b, (short)0, c,
                                               false, false);
}

// Async copy: 16 bytes global -> LDS per active lane, tracked by ASYNCcnt.
__device__ __forceinline__ void async_copy_b128(unsigned lds_byte_off,
                                                const float* src) {
  asm volatile("global_load_async_to_lds_b128 %0, %1, off"
               :
               : "v"(lds_byte_off), "v"(src)
               : "memory");
}
#define WAIT_ASYNCCNT(n) asm volatile("s_wait_asynccnt " #n ::: "memory")

// Low 32 bits of a generic LDS pointer == LDS byte offset (shared aperture).
__device__ __forceinline__ unsigned lds_addr(const void* p) {
  return (unsigned)(unsigned long long)p;
}

// Problem constants
#define BB 8
#define TT 2048
#define DD 1024
#define PP 512
#define SS 256

// ---------------------------------------------------------------------------
// Generic batched (optionally row-gathered) GEMM:
//   C[b][m][n] = sum_k A[b][g(m)][k] * W[k][n]        (lda == K, row-major)
// Block = 128 threads (4 waves); tile 16x64 (wave = one 16x16 WMMA tile).
// Double-buffered async global->LDS staging of A (16x32) and W (32x64).
// ---------------------------------------------------------------------------
#define KSTEP 32

__global__ __launch_bounds__(128)
void gemm_wmma_kernel(const float* __restrict__ A, long strideAb,
                      const int* __restrict__ gidx, int gstride,
                      const float* __restrict__ W,
                      float* __restrict__ C, long strideCb,
                      int M, int N, int K) {
  __shared__ float As[2][16][KSTEP];   // 2 x 2 KB
  __shared__ float Ws[2][KSTEP][64];   // 2 x 8 KB

  const int tid   = threadIdx.x;
  const int lane  = tid & 31;
  const int wave  = tid >> 5;
  const int l15   = lane & 15;
  const int khalf = (lane >> 4) * 2;   // 0 or 2 (K split across lane halves)
  const int m0    = blockIdx.x * 16;
  const int nblk  = blockIdx.y * 64;
  const int n0    = nblk + wave * 16;
  const int b     = blockIdx.z;

  const float* Ab = A + (long)b * strideAb;
  float*       Cb = C + (long)b * strideCb;

  // --- per-thread async staging coordinates -------------------------------
  // A tile: 16 rows x 32 cols; lane -> (row = tid/8, cols = (tid%8)*4 .. +3)
  const int ar = tid >> 3;
  const int ac = (tid & 7) * 4;
  const int arow = gidx ? gidx[b * gstride + m0 + ar] : (m0 + ar);
  const float* aSrc = Ab + (long)arow * K + ac;   // + k0 per slab
  // W slab: 32 rows x 64 cols = 2048 floats; 4 chunks of 512 floats.
  const int wr0 = (tid * 4) >> 6;                 // 0..7
  const int wc0 = (tid * 4) & 63;
  unsigned aDst[2], wDst[2][4];
#pragma unroll
  for (int p = 0; p < 2; p++) {
    aDst[p] = lds_addr(&As[p][ar][ac]);
#pragma unroll
    for (int j = 0; j < 4; j++)
      wDst[p][j] = lds_addr(&Ws[p][wr0 + 8 * j][wc0]);
  }

  // 5 async ops per wave per slab (1 A + 4 W).
  auto issue_slab = [&](int k0, int p) {
    async_copy_b128(aDst[p], aSrc + k0);
#pragma unroll
    for (int j = 0; j < 4; j++)
      async_copy_b128(wDst[p][j],
                      W + (long)(k0 + wr0 + 8 * j) * N + nblk + wc0);
  };

  v8f acc = {};
  const int nslab = K / KSTEP;
  issue_slab(0, 0);

  for (int i = 0; i < nslab; ++i) {
    const int p = i & 1;
    if (i + 1 < nslab) {
      issue_slab((i + 1) * KSTEP, p ^ 1);
      WAIT_ASYNCCNT(0x5);   // previous slab's 5 ops retired; next 5 in flight
    } else {
      WAIT_ASYNCCNT(0x0);
    }
    __syncthreads();        // all waves' portions of slab i are in LDS

#pragma unroll 4
    for (int kk = 0; kk < KSTEP; kk += 4) {
      v2f a, bb;
      // A frag (16x4): lanes 0-15 K={kk,kk+1}; lanes 16-31 K={kk+2,kk+3}
      a.x = As[p][l15][kk + khalf];
      a.y = As[p][l15][kk + khalf + 1];
      // B frag (4x16): v0 rows K={kk,kk+2}; v1 rows K={kk+1,kk+3}
      bb.x = Ws[p][kk + khalf][wave * 16 + l15];
      bb.y = Ws[p][kk + khalf + 1][wave * 16 + l15];
      acc = wmma_f32x4(a, bb, acc);
    }
    __syncthreads();        // safe to overwrite buffer p next-next slab
  }

  // C/D layout: VGPR r, lane<16 -> row r, col lane; lane>=16 -> row r+8.
#pragma unroll
  for (int r = 0; r < 8; r++) {
    int row = r + 8 * (lane >> 4);
    Cb[(long)(m0 + row) * N + n0 + l15] = acc[r];
  }
}

// ---------------------------------------------------------------------------
// scores[t] = sigmoid( sum_h relu(H[t][h]) * w2[h] ), one wave32 per token.
// ---------------------------------------------------------------------------
__global__ __launch_bounds__(256)
void score_reduce_kernel(const float* __restrict__ H,
                         const float* __restrict__ w2,
                         float* __restrict__ scores, int ntok) {
  int gw   = (blockIdx.x * blockDim.x + threadIdx.x) >> 5;
  int lane = threadIdx.x & 31;
  if (gw >= ntok) return;
  float s = 0.f;
  for (int h = lane; h < SS; h += 32) {
    float x = H[(long)gw * SS + h];
    s += fmaxf(x, 0.f) * w2[h];
  }
  for (int o = 16; o > 0; o >>= 1) s += __shfl_xor(s, o, 32);
  if (lane == 0) scores[gw] = 1.f / (1.f + __expf(-s));
}

// ---------------------------------------------------------------------------
// Per-batch bitonic sort (2048 elems) matching stable argsort of -scores:
// descending score, ascending original index on ties. Keeps first 512.
// ---------------------------------------------------------------------------
__global__ __launch_bounds__(256)
void sort_kernel(const float* __restrict__ scores,
                 int* __restrict__ sidx, float* __restrict__ sscore) {
  __shared__ float s[TT];
  __shared__ int   id[TT];
  const int b = blockIdx.x, tid = threadIdx.x;
  for (int i = tid; i < TT; i += 256) { s[i] = scores[b * TT + i]; id[i] = i; }
  __syncthreads();
  for (int k = 2; k <= TT; k <<= 1) {
    for (int j = k >> 1; j > 0; j >>= 1) {
      for (int i = tid; i < TT; i += 256) {
        int l = i ^ j;
        if (l > i) {
          float si = s[i], sl = s[l];
          int   ii = id[i], il = id[l];
          bool before = (si > sl) || (si == sl && ii < il);
          bool up = ((i & k) == 0);
          if (up != before) { s[i] = sl; s[l] = si; id[i] = il; id[l] = ii; }
        }
      }
      __syncthreads();
    }
  }
  for (int i = tid; i < PP; i += 256) {
    sidx[b * PP + i] = id[i];
    sscore[b * PP + i] = s[i];
  }
}

// ---------------------------------------------------------------------------
// Sequential pool update scan (one block per batch, uniform control flow).
// Writes pool/priorities/counts directly into their d_out regions.
// ---------------------------------------------------------------------------
__global__ __launch_bounds__(256)
void update_kernel(const float* __restrict__ pool_in,
                   const float* __restrict__ pri_in,
                   const int* __restrict__ counts_in,
                   const float* __restrict__ sscore,
                   const float* __restrict__ summaries,
                   float* __restrict__ pool_out,
                   float* __restrict__ pri_out,
                   int* __restrict__ counts_out) {
  const int b = blockIdx.x, tid = threadIdx.x;
  __shared__ float pri[PP];
  __shared__ float rv[256];
  __shared__ int   ri[256];

  for (int i = tid; i < PP; i += 256) pri[i] = pri_in[b * PP + i];
  for (long i = tid; i < (long)PP * SS; i += 256)
    pool_out[(long)b * PP * SS + i] = pool_in[(long)b * PP * SS + i];
  __syncthreads();

  int cnt = counts_in[b];  // uniform scalar, tracked redundantly in all lanes

  for (int slot = 0; slot < PP; slot++) {
    float imp = sscore[b * PP + slot];
    bool  has = imp > 0.5f;  // TAU1
    const float* summ = summaries + ((long)b * PP + slot) * SS;

    bool add = has && (cnt < PP);
    if (add) {
      pool_out[((long)b * PP + cnt) * SS + tid] = summ[tid];
      if (tid == 0) pri[cnt] = imp;
    }
    __syncthreads();
    if (add) cnt++;

    bool replace = has && (cnt >= PP);
    if (replace) {
      // first-min argmin over pri[0..511]
      float v0 = pri[tid], v1 = pri[tid + 256];
      if (v0 <= v1) { rv[tid] = v0; ri[tid] = tid; }
      else          { rv[tid] = v1; ri[tid] = tid + 256; }
      __syncthreads();
      for (int off = 128; off > 0; off >>= 1) {
        if (tid < off) {
          float a = rv[tid], c = rv[tid + off];
          int ia = ri[tid], ic = ri[tid + off];
          if (c < a || (c == a && ic < ia)) { rv[tid] = c; ri[tid] = ic; }
        }
        __syncthreads();
      }
      int midx = ri[0];
      float mval = rv[0];
      __syncthreads();
      if (imp > mval) {
        pool_out[((long)b * PP + midx) * SS + tid] = summ[tid];
        if (tid == 0) pri[midx] = imp;
      }
      __syncthreads();
    }
  }

  for (int i = tid; i < PP; i += 256) pri_out[b * PP + i] = pri[i];
  if (tid == 0) counts_out[b] = cnt;
}

// ---------------------------------------------------------------------------
// Fused attention: logits = q·k^T/sqrt(S) (WMMA) -> masked softmax (LDS) ->
// retrieved = probs·V (WMMA). One block per (batch, 16-token tile), 8 waves.
// ---------------------------------------------------------------------------
__global__ __launch_bounds__(256)
void attn_kernel(const float* __restrict__ q,
                 const float* __restrict__ kmat,
                 const float* __restrict__ vmat,
                 const int* __restrict__ counts,
                 float* __restrict__ out) {
  __shared__ float Qs[16][SS];   // 16 KB
  __shared__ float Lg[16][PP];   // 32 KB
  const int b = blockIdx.y, m0 = blockIdx.x * 16;
  const int tid = threadIdx.x, lane = tid & 31, wave = tid >> 5;
  const int l15 = lane & 15, khalf = (lane >> 4) * 2;
  const float* qb = q + ((long)b * TT + m0) * SS;
  const float* kb = kmat + (long)b * PP * SS;
  const float* vb = vmat + (long)b * PP * DD;

  for (int i = tid; i < 16 * SS; i += 256) Qs[i >> 8][i & 255] = qb[i];
  __syncthreads();

  // Logits: 32 p-tiles of 16, 4 per wave. B = K^T, (s,p) = kb[p*S + s].
  const float scale = 0.0625f;  // 1/sqrt(256)
#pragma unroll 1
  for (int t = 0; t < 4; t++) {
    int p0 = (wave + 8 * t) * 16;
    v8f acc = {};
#pragma unroll 4
    for (int s = 0; s < SS; s += 4) {
      v2f a, bv;
      a.x  = Qs[l15][s + khalf];
      a.y  = Qs[l15][s + khalf + 1];
      bv.x = kb[(long)(p0 + l15) * SS + s + khalf];
      bv.y = kb[(long)(p0 + l15) * SS + s + khalf + 1];
      acc = wmma_f32x4(a, bv, acc);
    }
#pragma unroll
    for (int r = 0; r < 8; r++)
      Lg[r + 8 * (lane >> 4)][p0 + l15] = acc[r] * scale;
  }
  __syncthreads();

  // Masked, numerically-stable softmax; count==0 -> zeros (nan_to_num).
  int cnt = counts[b];
  if (tid < 16) {
    float mx = -INFINITY;
    for (int p = 0; p < cnt; p++) mx = fmaxf(mx, Lg[tid][p]);
    float sum = 0.f;
    for (int p = 0; p < PP; p++) {
      float e = (p < cnt) ? __expf(Lg[tid][p] - mx) : 0.f;
      Lg[tid][p] = e;
      sum += e;
    }
    float inv = (cnt > 0 && sum > 0.f) ? 1.f / sum : 0.f;
    for (int p = 0; p < PP; p++) Lg[tid][p] *= inv;
  }
  __syncthreads();

  // retrieved = probs [16,512] @ V [512,1024]; 64 n-tiles, 8 per wave.
#pragma unroll 1
  for (int t = 0; t < 8; t++) {
    int n0 = (wave + 8 * t) * 16;
    v8f acc = {};
#pragma unroll 4
    for (int s = 0; s < PP; s += 4) {
      v2f a, bv;
      a.x  = Lg[l15][s + khalf];
      a.y  = Lg[l15][s + khalf + 1];
      bv.x = vb[(long)(s + khalf) * DD + n0 + l15];
      bv.y = vb[(long)(s + khalf + 1) * DD + n0 + l15];
      acc = wmma_f32x4(a, bv, acc);
    }
#pragma unroll
    for (int r = 0; r < 8; r++) {
      int row = r + 8 * (lane >> 4);
      out[((long)b * TT + m0 + row) * DD + n0 + l15] = acc[r];
    }
  }
}

// ---------------------------------------------------------------------------
extern "C" void kernel_launch(void* const* d_in, const int* in_sizes, int n_in,
                              void* d_out, int out_size, void* d_ws,
                              size_t ws_size, hipStream_t stream) {
  (void)in_sizes; (void)n_in; (void)out_size; (void)ws_size;
  const float* tokens = (const float*)d_in[0];
  const float* pool   = (const float*)d_in[1];
  const float* prio   = (const float*)d_in[2];
  const int*   counts = (const int*)d_in[3];
  const float* w1     = (const float*)d_in[4];
  const float* w2     = (const float*)d_in[5];
  const float* w_sum  = (const float*)d_in[6];
  const float* wq     = (const float*)d_in[7];
  const float* wk     = (const float*)d_in[8];
  const float* wv     = (const float*)d_in[9];

  // Output layout: retrieved | pool | priorities | counts
  float* retrieved  = (float*)d_out;
  float* pool_out   = retrieved + (long)BB * TT * DD;
  float* pri_out    = pool_out + (long)BB * PP * SS;
  int*   counts_out = (int*)(pri_out + (long)BB * PP);

  // Workspace layout (q aliases H: H is dead before q is produced).
  char* ws = (char*)d_ws;
  float* H      = (float*)ws;                                   // 16 MB
  float* q      = H;                                            // alias
  size_t off    = (size_t)BB * TT * SS * sizeof(float);
  float* scores = (float*)(ws + off); off += (size_t)BB * TT * sizeof(float);
  int*   sidx   = (int*)(ws + off);   off += (size_t)BB * PP * sizeof(int);
  float* sscore = (float*)(ws + off); off += (size_t)BB * PP * sizeof(float);
  float* summ   = (float*)(ws + off); off += (size_t)BB * PP * SS * sizeof(float);
  float* kws    = (float*)(ws + off); off += (size_t)BB * PP * SS * sizeof(float);
  float* vws    = (float*)(ws + off); off += (size_t)BB * PP * DD * sizeof(float);

  const int M_tok = BB * TT;  // 16384

  // 1) H = tokens @ w1   [16384,1024]x[1024,256]
  gemm_wmma_kernel<<<dim3(M_tok / 16, SS / 64, 1), 128, 0, stream>>>(
      tokens, 0, nullptr, 0, w1, H, 0, M_tok, SS, DD);
  // 2) scores = sigmoid(relu(H)·w2)
  score_reduce_kernel<<<dim3(M_tok / 8), 256, 0, stream>>>(H, w2, scores, M_tok);
  // 3) per-batch descending stable sort
  sort_kernel<<<dim3(BB), 256, 0, stream>>>(scores, sidx, sscore);
  // 4) summaries[b] = tokens[b, sidx, :] @ w_sum
  gemm_wmma_kernel<<<dim3(PP / 16, SS / 64, BB), 128, 0, stream>>>(
      tokens, (long)TT * DD, sidx, PP, w_sum, summ, (long)PP * SS, PP, SS, DD);
  // 5) sequential pool update (writes pool/pri/counts outputs)
  update_kernel<<<dim3(BB), 256, 0, stream>>>(pool, prio, counts, sscore, summ,
                                              pool_out, pri_out, counts_out);
  // 6) q = tokens @ wq (into aliased H region — H already consumed)
  gemm_wmma_kernel<<<dim3(M_tok / 16, SS / 64, 1), 128, 0, stream>>>(
      tokens, 0, nullptr, 0, wq, q, 0, M_tok, SS, DD);
  // 7) k[b] = pool_new[b] @ wk   [512,256]x[256,256]
  gemm_wmma_kernel<<<dim3(PP / 16, SS / 64, BB), 128, 0, stream>>>(
      pool_out, (long)PP * SS, nullptr, 0, wk, kws, (long)PP * SS, PP, SS, SS);
  // 8) v[b] = pool_new[b] @ wv   [512,256]x[256,1024]
  gemm_wmma_kernel<<<dim3(PP / 16, DD / 64, BB), 128, 0, stream>>>(
      pool_out, (long)PP * SS, nullptr, 0, wv, vws, (long)PP * DD, PP, DD, SS);
  // 9) fused masked attention -> retrieved
  attn_kernel<<<dim3(TT / 16, BB), 256, 0, stream>>>(q, kws, vws, counts_out,
                                                     retrieved);
}